// SphericalPreprocessor_57269093925344
// MI455X (gfx1250) — compile-verified
//
#include <hip/hip_runtime.h>
#include <hip/hip_bf16.h>
#include <math.h>

// ---------------------------------------------------------------------------
// SphericalPreprocessor for MI455X (gfx1250, wave32).
//
// dots = u(N x 3) @ dirs.T(3 x V) streamed through V_WMMA_F32_16X16X4_F32
// (16 points x 16 vertices per WMMA, K=4 zero-padded). Inner loop is manually
// unrolled x2: two b64 loads issued together, two WMMAs, two INDEPENDENT
// branchless top-3 chains (merged after the loop) -> load latency hidden by
// the cndmask chain, halved serial dependency depth. Split-K across 8
// workgroups per tile for device-wide occupancy; global merge kernel.
// ---------------------------------------------------------------------------

typedef float v2f __attribute__((ext_vector_type(2)));
typedef float v8f __attribute__((ext_vector_type(8)));

#define NSPLIT 8
#define NLEV 5

struct Params {
    const float* pos[NLEV];   // vertex arrays
    int  V[NLEV];             // vertex counts
    int  tileStart[NLEV + 1]; // prefix of ceil(V/16)
    int  vertStart[NLEV + 1]; // prefix of V
    long featOff[NLEV];       // output offsets (floats)
    long edgeOff[NLEV];
};

__global__ void zero_ws_kernel(float* ws) {
    if (threadIdx.x < 4) ws[threadIdx.x] = 0.0f;
}

// Block partial sums + atomicAdd -> ws[0..2] = sum(points, axis=0)
__global__ void __launch_bounds__(256)
com_sum_kernel(const float* __restrict__ pts, int n, float* __restrict__ sums) {
    __shared__ float sx[256], sy[256], sz[256];
    const int tid = threadIdx.x;
    float ax = 0.f, ay = 0.f, az = 0.f;
    for (int i = blockIdx.x * blockDim.x + tid; i < n; i += gridDim.x * blockDim.x) {
        ax += pts[i * 3 + 0];
        ay += pts[i * 3 + 1];
        az += pts[i * 3 + 2];
    }
    sx[tid] = ax; sy[tid] = ay; sz[tid] = az;
    __syncthreads();
    for (int s = 128; s > 0; s >>= 1) {
        if (tid < s) { sx[tid] += sx[tid + s]; sy[tid] += sy[tid + s]; sz[tid] += sz[tid + s]; }
        __syncthreads();
    }
    if (tid == 0) {
        atomicAdd(&sums[0], sx[0]);
        atomicAdd(&sums[1], sy[0]);
        atomicAdd(&sums[2], sz[0]);
    }
}

// u = normalize(p - com) stored as float4 (x,y,z,0); zero K-pad keeps the
// 16x16x4 WMMA exact for the 3-element dot product.
__global__ void __launch_bounds__(256)
compute_u_kernel(const float* __restrict__ pts, const float* __restrict__ sums,
                 float* __restrict__ u4, int n) {
    const int i = blockIdx.x * blockDim.x + threadIdx.x;
    if (i >= n) return;
    const float inv = 1.0f / (float)n;
    const float x = pts[i * 3 + 0] - sums[0] * inv;
    const float y = pts[i * 3 + 1] - sums[1] * inv;
    const float z = pts[i * 3 + 2] - sums[2] * inv;
    const float r = 1.0f / fmaxf(sqrtf(x * x + y * y + z * z), 1e-12f);
    u4[i * 4 + 0] = x * r;
    u4[i * 4 + 1] = y * r;
    u4[i * 4 + 2] = z * r;
    u4[i * 4 + 3] = 0.0f;
}

// Branchless ordered top-3 insert. Invariant t0>=t1>=t2 => g0 -> g1 -> g2.
// Strict '>' keeps the earlier (lower) index on ties (jax.lax.top_k).
#define TOP3_INSERT(v, pidx, t0, t1, t2, i0, i1, i2)                     \
    {                                                                    \
        const bool g0 = (v) > (t0);                                      \
        const bool g1 = (v) > (t1);                                      \
        const bool g2 = (v) > (t2);                                      \
        const float o0 = (t0), o1 = (t1);                                \
        const int   k0 = (i0), k1 = (i1);                                \
        (t0) = g0 ? (v)    : (t0);                                       \
        (i0) = g0 ? (pidx) : (i0);                                       \
        (t1) = g0 ? o0 : (g1 ? (v)    : (t1));                           \
        (i1) = g0 ? k0 : (g1 ? (pidx) : (i1));                           \
        (t2) = g1 ? o1 : (g2 ? (v)    : (t2));                           \
        (i2) = g1 ? k1 : (g2 ? (pidx) : (i2));                           \
    }

// Insert with value/index tie-break (for merging independent streams).
#define TOP3_INSERT_TIE(v, id, b0, b1, b2, j0, j1, j2)                   \
    {                                                                    \
        const bool g0 = ((v) > (b0)) || ((v) == (b0) && (id) < (j0));    \
        const bool g1 = ((v) > (b1)) || ((v) == (b1) && (id) < (j1));    \
        const bool g2 = ((v) > (b2)) || ((v) == (b2) && (id) < (j2));    \
        const float o0 = (b0), o1 = (b1);                                \
        const int   k0 = (j0), k1 = (j1);                                \
        (b0) = g0 ? (v)  : (b0);                                         \
        (j0) = g0 ? (id) : (j0);                                         \
        (b1) = g0 ? o0 : (g1 ? (v)  : (b1));                             \
        (j1) = g0 ? k0 : (g1 ? (id) : (j1));                             \
        (b2) = g1 ? o1 : (g2 ? (v)  : (b2));                             \
        (j2) = g1 ? k1 : (g2 ? (id) : (j2));                             \
    }

// blockIdx.x = global tile t (all levels fused), blockIdx.y = point-split s.
// A layout (ISA 7.12.2, f32 16x4): lanes 0-15 hold (K0,K1)=(u.x,u.y) of point
// M=lane; lanes 16-31 hold (K2,K3)=(u.z,0) -> lane-fixed b64 load, no selects.
// D layout: column N = lane%16 (vertex), VGPR r = row M=r (+8 for hi lanes).
__global__ void __launch_bounds__(256)
topk_partial_kernel(const float* __restrict__ u4, Params p, int n,
                    float* __restrict__ pVal, int* __restrict__ pIdx) {
    const int t = blockIdx.x;
    int lev = 0;
    while (lev < NLEV - 1 && t >= p.tileStart[lev + 1]) ++lev;
    const int tileBase = (t - p.tileStart[lev]) * 16;
    const int V = p.V[lev];
    const float* __restrict__ pos = p.pos[lev];

    const int s    = blockIdx.y;
    const int tid  = threadIdx.x;
    const int lane = tid & 31;
    const int wave = tid >> 5;
    const int col  = lane & 15;
    const bool hi  = lane >= 16;
    const int vidx = tileBase + col;

    // B operand: normalized vertex direction (zero-padded past V).
    float dx = 0.f, dy = 0.f, dz = 0.f;
    if (vidx < V) {
        const float px = pos[vidx * 3 + 0];
        const float py = pos[vidx * 3 + 1];
        const float pz = pos[vidx * 3 + 2];
        const float r = 1.0f / fmaxf(sqrtf(px * px + py * py + pz * pz), 1e-12f);
        dx = px * r; dy = py * r; dz = pz * r;
    }
    v2f b;
    b[0] = hi ? dz : dx;
    b[1] = hi ? 0.f : dy;

    // Two independent top-3 streams (halved serial cndmask depth).
    float ta0 = -__builtin_inff(), ta1 = -__builtin_inff(), ta2 = -__builtin_inff();
    int   ia0 = 0x7fffffff, ia1 = 0x7fffffff, ia2 = 0x7fffffff;
    float tb0 = -__builtin_inff(), tb1 = -__builtin_inff(), tb2 = -__builtin_inff();
    int   ib0 = 0x7fffffff, ib1 = 0x7fffffff, ib2 = 0x7fffffff;

    const float* lanePtr = u4 + col * 4 + (hi ? 2 : 0); // lane-fixed component pair
    const int   laneMOff = hi ? 8 : 0;

    const int nblocks = n >> 4;
    const int chunk   = (nblocks + NSPLIT - 1) / NSPLIT;
    const int beg     = s * chunk;
    const int end     = (beg + chunk < nblocks) ? beg + chunk : nblocks;

    int pb = beg + wave;
    // Main loop: 32 points per iteration, two independent load->WMMA->top3
    // streams. Both b64 loads issue before the first WMMA consumes its data
    // (in-order load returns => WMMA0 needs only loadcnt<=1); the ~35-VALU
    // chain A covers the in-flight latency of load1.
    for (; pb + 8 < end; pb += 16) {
        const v2f a0 = *(const v2f*)(lanePtr + (size_t)pb * 64);
        const v2f a1 = *(const v2f*)(lanePtr + (size_t)(pb + 8) * 64);
        v8f c0 = {};
        v8f d0 = __builtin_amdgcn_wmma_f32_16x16x4_f32(
            false, a0, false, b, (short)0, c0, false, false);
        const int poff0 = (pb << 4) + laneMOff;
#pragma unroll
        for (int r = 0; r < 8; ++r) {
            const float v = d0[r];
            const int pidx = poff0 + r;
            TOP3_INSERT(v, pidx, ta0, ta1, ta2, ia0, ia1, ia2);
        }
        v8f c1 = {};
        v8f d1 = __builtin_amdgcn_wmma_f32_16x16x4_f32(
            false, a1, false, b, (short)0, c1, false, false);
        const int poff1 = ((pb + 8) << 4) + laneMOff;
#pragma unroll
        for (int r = 0; r < 8; ++r) {
            const float v = d1[r];
            const int pidx = poff1 + r;
            TOP3_INSERT(v, pidx, tb0, tb1, tb2, ib0, ib1, ib2);
        }
    }
    // Epilogue (trip count generality; dead for N=131072 / NSPLIT=8).
    for (; pb < end; pb += 8) {
        const v2f a0 = *(const v2f*)(lanePtr + (size_t)pb * 64);
        v8f c0 = {};
        v8f d0 = __builtin_amdgcn_wmma_f32_16x16x4_f32(
            false, a0, false, b, (short)0, c0, false, false);
        const int poff0 = (pb << 4) + laneMOff;
#pragma unroll
        for (int r = 0; r < 8; ++r) {
            const float v = d0[r];
            const int pidx = poff0 + r;
            TOP3_INSERT(v, pidx, ta0, ta1, ta2, ia0, ia1, ia2);
        }
    }
    // Merge stream B into stream A (tie-break by index: order-independent).
    TOP3_INSERT_TIE(tb0, ib0, ta0, ta1, ta2, ia0, ia1, ia2);
    TOP3_INSERT_TIE(tb1, ib1, ta0, ta1, ta2, ia0, ia1, ia2);
    TOP3_INSERT_TIE(tb2, ib2, ta0, ta1, ta2, ia0, ia1, ia2);

    __shared__ float sV[256][3];
    __shared__ int   sI[256][3];
    sV[tid][0] = ta0; sV[tid][1] = ta1; sV[tid][2] = ta2;
    sI[tid][0] = ia0; sI[tid][1] = ia1; sI[tid][2] = ia2;
    __syncthreads();

    // Merge 16 lanes x 3 candidates per vertex -> per-WG top-3 partial.
    if (tid < 16 && (tileBase + tid) < V) {
        const int j = tid;
        float b0 = -__builtin_inff(), b1 = -__builtin_inff(), b2 = -__builtin_inff();
        int   j0 = 0x7fffffff, j1 = 0x7fffffff, j2 = 0x7fffffff;
        for (int w = 0; w < 8; ++w) {
            for (int h = 0; h < 2; ++h) {
                const int L = w * 32 + h * 16 + j;
                for (int k = 0; k < 3; ++k) {
                    const float v = sV[L][k];
                    const int  id = sI[L][k];
                    TOP3_INSERT_TIE(v, id, b0, b1, b2, j0, j1, j2);
                }
            }
        }
        const int base = ((t * NSPLIT + s) * 16 + j) * 3;
        pVal[base + 0] = b0; pVal[base + 1] = b1; pVal[base + 2] = b2;
        pIdx[base + 0] = j0; pIdx[base + 1] = j1; pIdx[base + 2] = j2;
    }
}

// One (32-thread) block per tile: merge NSPLIT x 3 partials per vertex,
// gather intensity, emit feats = [mean_inten, pos.xyz].
__global__ void __launch_bounds__(32)
topk_merge_kernel(const float* __restrict__ pVal, const int* __restrict__ pIdx,
                  const float* __restrict__ inten, Params p,
                  float* __restrict__ out) {
    const int t = blockIdx.x;
    int lev = 0;
    while (lev < NLEV - 1 && t >= p.tileStart[lev + 1]) ++lev;
    const int tileBase = (t - p.tileStart[lev]) * 16;
    const int V = p.V[lev];
    const int j = threadIdx.x;
    if (j >= 16 || tileBase + j >= V) return;

    float b0 = -__builtin_inff(), b1 = -__builtin_inff(), b2 = -__builtin_inff();
    int   j0 = 0x7fffffff, j1 = 0x7fffffff, j2 = 0x7fffffff;
    for (int s = 0; s < NSPLIT; ++s) {
        const int base = ((t * NSPLIT + s) * 16 + j) * 3;
        for (int k = 0; k < 3; ++k) {
            const float v = pVal[base + k];
            const int  id = pIdx[base + k];
            TOP3_INSERT_TIE(v, id, b0, b1, b2, j0, j1, j2);
        }
    }
    const float* pos = p.pos[lev];
    float* outFeats = out + p.featOff[lev];
    const float m = (inten[j0] + inten[j1] + inten[j2]) * (1.0f / 3.0f);
    const int vo = tileBase + j;
    outFeats[vo * 4 + 0] = m;
    outFeats[vo * 4 + 1] = pos[vo * 3 + 0];
    outFeats[vo * 4 + 2] = pos[vo * 3 + 1];
    outFeats[vo * 4 + 3] = pos[vo * 3 + 2];
}

// All-levels V x V kNN (k=6, self excluded) + edge emission. Layout matches
// flatten(concat([ei, ei[::-1]], axis=1)) = [src | knn | knn | src].
__global__ void __launch_bounds__(256)
knn_edges_kernel(Params p, float* __restrict__ out) {
    const int gv = blockIdx.x * blockDim.x + threadIdx.x;
    if (gv >= p.vertStart[NLEV]) return;
    int lev = 0;
    while (lev < NLEV - 1 && gv >= p.vertStart[lev + 1]) ++lev;
    const int i = gv - p.vertStart[lev];
    const int V = p.V[lev];
    const float* __restrict__ pos = p.pos[lev];
    float* __restrict__ outE = out + p.edgeOff[lev];

    const float px = pos[i * 3 + 0], py = pos[i * 3 + 1], pz = pos[i * 3 + 2];
    const float sqi = px * px + py * py + pz * pz;
    float bd[6]; int bi[6];
#pragma unroll
    for (int k = 0; k < 6; ++k) { bd[k] = __builtin_inff(); bi[k] = 0x7fffffff; }
    for (int j = 0; j < V; ++j) {
        if (j == i) continue;                    // self d2 == 0 exactly -> dropped row
        const float qx = pos[j * 3 + 0], qy = pos[j * 3 + 1], qz = pos[j * 3 + 2];
        const float sqj = qx * qx + qy * qy + qz * qz;
        const float dot = px * qx + py * qy + pz * qz;
        const float d2 = sqi + sqj - 2.0f * dot; // same formula as reference
        if (d2 < bd[5]) {                        // strict < keeps earlier j on ties
            bd[5] = d2; bi[5] = j;
            for (int k = 5; k > 0 && bd[k] < bd[k - 1]; --k) {
                const float tv = bd[k]; bd[k] = bd[k - 1]; bd[k - 1] = tv;
                const int ti = bi[k]; bi[k] = bi[k - 1]; bi[k - 1] = ti;
            }
        }
    }
    const int E = V * 6;
#pragma unroll
    for (int s = 0; s < 6; ++s) {
        const int e = i * 6 + s;
        outE[e]         = (float)i;       // row0 first half : src
        outE[E + e]     = (float)bi[s];   // row0 second half: knn
        outE[2 * E + e] = (float)bi[s];   // row1 first half : knn
        outE[3 * E + e] = (float)i;       // row1 second half: src
    }
}

extern "C" void kernel_launch(void* const* d_in, const int* in_sizes, int n_in,
                              void* d_out, int out_size, void* d_ws, size_t ws_size,
                              hipStream_t stream) {
    // dict order: points_xyz, intensity, vertices (5 arrays), knn_k, neighbor_k
    const float* pts   = (const float*)d_in[0];
    const float* inten = (const float*)d_in[1];
    const int N = in_sizes[0] / 3;

    Params p;
    long off = 0;
    int tiles = 0, verts = 0;
    for (int l = 0; l < NLEV; ++l) {
        p.pos[l] = (const float*)d_in[2 + l];
        p.V[l]   = in_sizes[2 + l] / 3;
        p.tileStart[l] = tiles;  tiles += (p.V[l] + 15) / 16;
        p.vertStart[l] = verts;  verts += p.V[l];
    }
    p.tileStart[NLEV] = tiles;
    p.vertStart[NLEV] = verts;
    for (int l = 0; l < NLEV; ++l) { p.featOff[l] = off; off += (long)p.V[l] * 4; }
    for (int l = 0; l < NLEV; ++l) { p.edgeOff[l] = off; off += (long)p.V[l] * 24; }

    // Workspace: sums(16f) | u4 (N*4 f) | pVal (tiles*NSPLIT*16*3 f) | pIdx (same, int)
    float* ws   = (float*)d_ws;
    float* sums = ws;
    float* u4   = ws + 16;
    float* pVal = u4 + (size_t)N * 4;
    int*   pIdx = (int*)(pVal + (size_t)tiles * NSPLIT * 16 * 3);
    float* out  = (float*)d_out;

    zero_ws_kernel<<<1, 32, 0, stream>>>(sums);
    com_sum_kernel<<<256, 256, 0, stream>>>(pts, N, sums);
    compute_u_kernel<<<(N + 255) / 256, 256, 0, stream>>>(pts, sums, u4, N);

    // Split-K top-3: (tiles x NSPLIT) workgroups fill the device even for the
    // 12-vertex level; then one tiny merge block per tile.
    topk_partial_kernel<<<dim3(tiles, NSPLIT), 256, 0, stream>>>(u4, p, N, pVal, pIdx);
    topk_merge_kernel<<<tiles, 32, 0, stream>>>(pVal, pIdx, inten, p, out);

    knn_edges_kernel<<<(verts + 255) / 256, 256, 0, stream>>>(p, out);
}